// GNN_3289944948994
// MI455X (gfx1250) — compile-verified
//
#include <hip/hip_runtime.h>

#define N_NODES 100000
#define N_EDGES 1600000
#define D 128

typedef __attribute__((ext_vector_type(2))) float v2f;
typedef __attribute__((ext_vector_type(8))) float v8f;

// ---------------------------------------------------------------------------
// Native non-returning f32 atomic add (GLOBAL_ATOMIC_ADD_F32, no CAS loop).
// 64-bit vaddr form; tracked with STOREcnt, drained at s_endpgm.
// ---------------------------------------------------------------------------
__device__ __forceinline__ void atomic_add_f32_native(float* p, float v) {
    asm volatile("global_atomic_add_f32 %0, %1, off"
                 :: "v"(p), "v"(v)
                 : "memory");
}

// ---------------------------------------------------------------------------
// Degree: one thread per edge, atomic count into deg[dst]
// ---------------------------------------------------------------------------
__global__ void __launch_bounds__(256)
deg_kernel(const long long* __restrict__ dst, float* __restrict__ deg) {
    int e = blockIdx.x * 256 + threadIdx.x;
    if (e < N_EDGES) {
        int d = (int)dst[e];
        atomic_add_f32_native(&deg[d], 1.0f);
    }
}

// rdeg[n] = 1 / max(deg[n], 1)
__global__ void __launch_bounds__(256)
rdeg_kernel(const float* __restrict__ deg, float* __restrict__ rdeg) {
    int n = blockIdx.x * 256 + threadIdx.x;
    if (n < N_NODES) {
        rdeg[n] = 1.0f / fmaxf(deg[n], 1.0f);
    }
}

// ---------------------------------------------------------------------------
// Scatter-add: one wave (32 lanes) per edge; each lane moves float4 (4 floats)
// agg[dst, :] += x[src, :]    (agg is 51 MB -> L2 resident, atomics hit L2)
// Grid: N_EDGES/8 blocks of 256 threads (8 waves/block) -> exactly E waves.
// ---------------------------------------------------------------------------
__global__ void __launch_bounds__(256)
scatter_add_kernel(const float* __restrict__ xin,
                   const long long* __restrict__ src,
                   const long long* __restrict__ dst,
                   float* __restrict__ agg) {
    int e = blockIdx.x * 8 + (threadIdx.x >> 5);
    int lane = threadIdx.x & 31;
    long long s = src[e];
    long long d = dst[e];
    const float4* xrow = (const float4*)(xin + (size_t)s * D);
    float4 v = xrow[lane];                    // global_load_b128
    float* a = agg + (size_t)d * D + lane * 4;
    atomic_add_f32_native(a + 0, v.x);
    atomic_add_f32_native(a + 1, v.y);
    atomic_add_f32_native(a + 2, v.z);
    atomic_add_f32_native(a + 3, v.w);
}

// ---------------------------------------------------------------------------
// Fused SAGE layer output:
//   out[n,:] = relu( (agg[n,:] * rdeg[n]) @ Wl + x[n,:] @ Wr + bias )
//
// One wave per 16x16 output tile, V_WMMA_F32_16X16X4_F32 over K=128 (32 steps,
// 2 accumulations per step -> 64 WMMAs/wave). Block = 256 threads = 8 waves
// covering all 8 column tiles of a 16-row node tile. Grid = N/16 = 6250.
//
// VGPR layouts (ISA 7.12.2, 16x16x4 f32):
//   A: lane<16 -> M=lane,   {v0,v1} = K = {k0,k0+1}
//      lane>=16-> M=lane-16,{v0,v1} = K = {k0+2,k0+3}
//   B: lane&15 = N, same K split as A (rows striped across lanes per VGPR)
//   C/D: vgpr r -> M = r + 8*(lane>>4), N = lane&15
// ---------------------------------------------------------------------------
__global__ void __launch_bounds__(256)
sage_gemm_kernel(const float* __restrict__ agg,
                 const float* __restrict__ xin,
                 const float* __restrict__ rdeg,
                 const float* __restrict__ Wl,
                 const float* __restrict__ Wr,
                 const float* __restrict__ bias,
                 float* __restrict__ out) {
    const int lane = threadIdx.x & 31;
    const int wave = threadIdx.x >> 5;   // 0..7 -> column tile
    const int nb   = wave * 16;          // column base
    const int mb   = blockIdx.x * 16;    // row (node) base

    const int mi = lane & 15;            // A row within tile / B,C column
    const int hi = lane >> 4;            // 0 or 1: K-pair select
    const int row = mb + mi;
    const int col = nb + mi;

    const float scale = rdeg[row];
    const float* aggRow = agg + (size_t)row * D;
    const float* xRow   = xin + (size_t)row * D;

    v8f c = {};
#pragma unroll 4
    for (int k0 = 0; k0 < D; k0 += 4) {
        const int ka = k0 + hi * 2;

        // A = normalized aggregate, B = Wl column panel
        v2f aA;
        aA.x = aggRow[ka]     * scale;
        aA.y = aggRow[ka + 1] * scale;
        v2f bL;
        bL.x = Wl[(size_t)ka * D + col];
        bL.y = Wl[(size_t)(ka + 1) * D + col];
        c = __builtin_amdgcn_wmma_f32_16x16x4_f32(
                false, aA, false, bL, (short)0, c, false, false);

        // A = node features, B = Wr column panel (same accumulator)
        v2f aX;
        aX.x = xRow[ka];
        aX.y = xRow[ka + 1];
        v2f bR;
        bR.x = Wr[(size_t)ka * D + col];
        bR.y = Wr[(size_t)(ka + 1) * D + col];
        c = __builtin_amdgcn_wmma_f32_16x16x4_f32(
                false, aX, false, bR, (short)0, c, false, false);
    }

    // Epilogue: bias + ReLU, store 16x16 tile.
    const float bn = bias[col];
    float* outBase = out + (size_t)(mb + 8 * hi) * D + col;
#pragma unroll
    for (int r = 0; r < 8; ++r) {
        float v = c[r] + bn;
        v = v > 0.0f ? v : 0.0f;
        outBase[(size_t)r * D] = v;
    }
}

// ---------------------------------------------------------------------------
// Host-side launch (graph-capture safe: only kernels + hipMemsetAsync)
// ---------------------------------------------------------------------------
extern "C" void kernel_launch(void* const* d_in, const int* in_sizes, int n_in,
                              void* d_out, int out_size, void* d_ws, size_t ws_size,
                              hipStream_t stream) {
    const float*     x   = (const float*)d_in[0];
    const long long* ei  = (const long long*)d_in[1];   // int64 [2, E]
    const float*     Wl1 = (const float*)d_in[2];
    const float*     Wr1 = (const float*)d_in[3];
    const float*     b1  = (const float*)d_in[4];
    const float*     Wl2 = (const float*)d_in[5];
    const float*     Wr2 = (const float*)d_in[6];
    const float*     b2  = (const float*)d_in[7];

    const long long* src = ei;
    const long long* dst = ei + N_EDGES;

    // Workspace layout: deg[N] | rdeg[N] | agg[N*D] | h1[N*D]  (~103 MB)
    float* deg  = (float*)d_ws;
    float* rdeg = deg + N_NODES;
    float* agg  = rdeg + N_NODES;
    float* h1   = agg + (size_t)N_NODES * D;

    const size_t aggBytes = (size_t)N_NODES * D * sizeof(float);

    const int eBlocks1t  = (N_EDGES + 255) / 256;   // 1 thread / edge
    const int eBlocksW   = N_EDGES / 8;             // 1 wave / edge (8 waves/block)
    const int nBlocks    = (N_NODES + 255) / 256;
    const int gemmBlocks = N_NODES / 16;            // 6250 (exact)

    // Degree + reciprocal (shared by both layers)
    hipMemsetAsync(deg, 0, N_NODES * sizeof(float), stream);
    deg_kernel<<<eBlocks1t, 256, 0, stream>>>(dst, deg);
    rdeg_kernel<<<nBlocks, 256, 0, stream>>>(deg, rdeg);

    // ---- Layer 1 ----
    hipMemsetAsync(agg, 0, aggBytes, stream);
    scatter_add_kernel<<<eBlocksW, 256, 0, stream>>>(x, src, dst, agg);
    sage_gemm_kernel<<<gemmBlocks, 256, 0, stream>>>(agg, x, rdeg,
                                                     Wl1, Wr1, b1, h1);

    // ---- Layer 2 ----
    hipMemsetAsync(agg, 0, aggBytes, stream);
    scatter_add_kernel<<<eBlocksW, 256, 0, stream>>>(h1, src, dst, agg);
    sage_gemm_kernel<<<gemmBlocks, 256, 0, stream>>>(agg, h1, rdeg,
                                                     Wl2, Wr2, b2, (float*)d_out);
}